// MoEMLP_52432960749602
// MI455X (gfx1250) — compile-verified
//
#include <hip/hip_runtime.h>
#include <hip/hip_bf16.h>
#include <math.h>

// ---------------- problem constants ----------------
#define TT   4096      // tokens (B*N)
#define DD   768       // model dim
#define HH   1536      // hidden dim
#define EE   64        // experts
#define KK   6         // top-k
#define CAPN 1536      // per-expert capacity
#define HC   384       // H-chunk held in LDS during fused expert MLP

typedef float v2f __attribute__((ext_vector_type(2)));
typedef float v8f __attribute__((ext_vector_type(8)));

static __device__ __forceinline__ v8f zero8() {
    v8f z = {0.f, 0.f, 0.f, 0.f, 0.f, 0.f, 0.f, 0.f};
    return z;
}

static __device__ __forceinline__ float gelu_exact(float v) {
    return 0.5f * v * (1.0f + erff(v * 0.70710678118654752440f));
}

// =====================================================================
// Kernel 1: gating. One wave (32 lanes) per token. Each lane owns
// experts {lane, lane+32}. Computes softmax over 64 logits, accumulates
// expert_prob sums, selects top-6 by iterative wave-argmax, and assigns
// capacity slots via atomics (slot order within an expert is irrelevant
// to the final scatter-add combine).
// =====================================================================
__global__ __launch_bounds__(256) void gate_topk_kernel(
    const float* __restrict__ x, const float* __restrict__ gate_w,
    int* __restrict__ counter, float* __restrict__ prob_sum,
    int* __restrict__ route_tok, float* __restrict__ route_sc)
{
    __shared__ float xs[8][DD];
    const int wave = threadIdx.x >> 5;
    const int lane = threadIdx.x & 31;
    const int tok  = blockIdx.x * 8 + wave;

    const float* xr = x + (size_t)tok * DD;
    for (int d = lane; d < DD; d += 32) xs[wave][d] = xr[d];
    __syncthreads();

    // two logits per lane
    const float* g0 = gate_w + (size_t)lane * DD;
    const float* g1 = gate_w + (size_t)(lane + 32) * DD;
    float s0 = 0.f, s1 = 0.f;
    for (int d = 0; d < DD; ++d) {
        float xv = xs[wave][d];
        s0 = fmaf(xv, g0[d], s0);
        s1 = fmaf(xv, g1[d], s1);
    }

    // softmax over 64 via wave32 butterfly
    float m = fmaxf(s0, s1);
    for (int off = 16; off; off >>= 1) m = fmaxf(m, __shfl_xor(m, off, 32));
    float e0 = expf(s0 - m), e1 = expf(s1 - m);
    float sum = e0 + e1;
    for (int off = 16; off; off >>= 1) sum += __shfl_xor(sum, off, 32);
    float inv = 1.0f / sum;
    float p0 = e0 * inv, p1 = e1 * inv;

    atomicAdd(&prob_sum[lane], p0);
    atomicAdd(&prob_sum[lane + 32], p1);

    // iterative top-6: wave argmax, ties -> lowest expert index
    float q0 = p0, q1 = p1;
    for (int k = 0; k < KK; ++k) {
        float v = q0; int ei = lane;
        if (q1 > v) { v = q1; ei = lane + 32; }
        for (int off = 16; off; off >>= 1) {
            float ov = __shfl_xor(v, off, 32);
            int   oe = __shfl_xor(ei, off, 32);
            if (ov > v || (ov == v && oe < ei)) { v = ov; ei = oe; }
        }
        // lane k records routing for the k-th pick
        if (lane == k) {
            int slot = atomicAdd(&counter[ei], 1);
            if (slot < CAPN) {
                route_tok[ei * CAPN + slot] = tok;
                route_sc [ei * CAPN + slot] = v;
            }
        }
        if (ei == lane)      q0 = -1.f;
        if (ei == lane + 32) q1 = -1.f;
    }
}

// =====================================================================
// Kernel 2: fused per-expert MLP tile. Block = (tile of 16 slots) x
// expert. GEMM1 (gathered x @ W1^T, gelu) chunked into 16xHC pieces in
// LDS, GEMM2 accumulated in VGPRs across chunks, scaled by gate score,
// atomically scattered into the routed accumulator.
// WMMA f32 16x16x4 operand mapping (wave32):
//   A 16x4 : row = lane&15, K pair = k + 2*(lane>>4)      (float2 load)
//   B 4x16 : col = lane&15, K pair = k + 2*(lane>>4)      (float2 load)
//   C 16x16: vgpr j -> M = j + 8*(lane>>4), N = lane&15
// =====================================================================
__global__ __launch_bounds__(128) void expert_mlp_kernel(
    const float* __restrict__ x,
    const float* __restrict__ e1w, const float* __restrict__ e1b,
    const float* __restrict__ e2w, const float* __restrict__ e2b,
    const int* __restrict__ counter,
    const int* __restrict__ route_tok, const float* __restrict__ route_sc,
    float* __restrict__ acc)
{
    __shared__ float hs[HC * 16];     // transposed h chunk: hs[hcol*16 + m]
    __shared__ int   t16[16];
    __shared__ float sc16[16];

    const int e    = blockIdx.y;
    const int row0 = blockIdx.x * 16;
    int cnt = counter[e];
    if (cnt > CAPN) cnt = CAPN;
    int nvalid = cnt - row0;
    if (nvalid <= 0) return;          // uniform per block
    if (nvalid > 16) nvalid = 16;

    const int tid = threadIdx.x;
    if (tid < 16) {
        int tk = 0; float s = 0.f;
        if (tid < nvalid) {
            tk = route_tok[e * CAPN + row0 + tid];
            s  = route_sc [e * CAPN + row0 + tid];
        }
        t16[tid] = tk; sc16[tid] = s;
    }
    __syncthreads();

    const int wave = tid >> 5;
    const int lane = tid & 31;
    const int half = lane >> 4;
    const int lrow = lane & 15;

    const float* arow = x + (size_t)t16[lrow] * DD + 2 * half;   // gathered A row stream
    const float* w1   = e1w + (size_t)e * HH * DD;
    const float* w2   = e2w + (size_t)e * DD * HH;
    const float* b1   = e1b + (size_t)e * HH;
    const float* b2   = e2b + (size_t)e * DD;

    v8f o[12];
    #pragma unroll
    for (int t = 0; t < 12; ++t) o[t] = zero8();

    for (int hc = 0; hc < HH / HC; ++hc) {
        __syncthreads();   // hs reuse guard
        // ---- GEMM1: this chunk's 16 x HC slice of h ----
        #pragma unroll
        for (int t = 0; t < 6; ++t) {
            const int nb_local = wave * 96 + t * 16;          // col base within chunk
            const int col      = hc * HC + nb_local + lrow;   // global H column (per lane)
            const float* brow  = w1 + (size_t)col * DD + 2 * half;
            v8f c = zero8();
            for (int k = 0; k < DD; k += 4) {
                v2f a = *(const v2f*)(arow + k);
                v2f b = *(const v2f*)(brow + k);
                c = __builtin_amdgcn_wmma_f32_16x16x4_f32(false, a, false, b,
                                                          (short)0, c, false, false);
            }
            const float bias = b1[col];
            #pragma unroll
            for (int j = 0; j < 8; ++j) {
                float v = gelu_exact(c[j] + bias);
                hs[(nb_local + lrow) * 16 + 8 * half + j] = v;
            }
        }
        __syncthreads();
        // ---- GEMM2 partial: accumulate over this HC chunk ----
        #pragma unroll
        for (int t = 0; t < 12; ++t) {
            const int ncol    = wave * 192 + t * 16 + lrow;   // global D column (per lane)
            const float* brow = w2 + (size_t)ncol * HH + hc * HC + 2 * half;
            v8f c = o[t];
            for (int kk = 0; kk < HC; kk += 4) {
                v2f a;
                a.x = hs[(kk + 2 * half)     * 16 + lrow];
                a.y = hs[(kk + 2 * half + 1) * 16 + lrow];
                v2f b = *(const v2f*)(brow + kk);
                c = __builtin_amdgcn_wmma_f32_16x16x4_f32(false, a, false, b,
                                                          (short)0, c, false, false);
            }
            o[t] = c;
        }
    }

    // ---- epilogue: bias, gate-score scale, scatter-add combine ----
    #pragma unroll
    for (int t = 0; t < 12; ++t) {
        const int ncol  = wave * 192 + t * 16 + lrow;
        const float bias = b2[ncol];
        #pragma unroll
        for (int j = 0; j < 8; ++j) {
            const int mrow = 8 * half + j;
            if (mrow < nvalid) {
                float v = (o[t][j] + bias) * sc16[mrow];
                atomicAdd(&acc[(size_t)t16[mrow] * DD + ncol], v);
            }
        }
    }
}

// =====================================================================
// Kernel 3: shared (dense) MLP over all tokens; same fused tiling.
// Runs after the expert kernel (stream-ordered), adds routed
// accumulator, writes final output.
// =====================================================================
__global__ __launch_bounds__(128) void shared_mlp_kernel(
    const float* __restrict__ x,
    const float* __restrict__ w1g, const float* __restrict__ b1g,
    const float* __restrict__ w2g, const float* __restrict__ b2g,
    const float* __restrict__ acc, float* __restrict__ out)
{
    __shared__ float hs[HC * 16];

    const int row0 = blockIdx.x * 16;
    const int tid  = threadIdx.x;
    const int wave = tid >> 5;
    const int lane = tid & 31;
    const int half = lane >> 4;
    const int lrow = lane & 15;

    const float* arow = x + (size_t)(row0 + lrow) * DD + 2 * half;

    v8f o[12];
    #pragma unroll
    for (int t = 0; t < 12; ++t) o[t] = zero8();

    for (int hc = 0; hc < HH / HC; ++hc) {
        __syncthreads();
        #pragma unroll
        for (int t = 0; t < 6; ++t) {
            const int nb_local = wave * 96 + t * 16;
            const int col      = hc * HC + nb_local + lrow;
            const float* brow  = w1g + (size_t)col * DD + 2 * half;
            v8f c = zero8();
            for (int k = 0; k < DD; k += 4) {
                v2f a = *(const v2f*)(arow + k);
                v2f b = *(const v2f*)(brow + k);
                c = __builtin_amdgcn_wmma_f32_16x16x4_f32(false, a, false, b,
                                                          (short)0, c, false, false);
            }
            const float bias = b1g[col];
            #pragma unroll
            for (int j = 0; j < 8; ++j) {
                float v = gelu_exact(c[j] + bias);
                hs[(nb_local + lrow) * 16 + 8 * half + j] = v;
            }
        }
        __syncthreads();
        #pragma unroll
        for (int t = 0; t < 12; ++t) {
            const int ncol    = wave * 192 + t * 16 + lrow;
            const float* brow = w2g + (size_t)ncol * HH + hc * HC + 2 * half;
            v8f c = o[t];
            for (int kk = 0; kk < HC; kk += 4) {
                v2f a;
                a.x = hs[(kk + 2 * half)     * 16 + lrow];
                a.y = hs[(kk + 2 * half + 1) * 16 + lrow];
                v2f b = *(const v2f*)(brow + kk);
                c = __builtin_amdgcn_wmma_f32_16x16x4_f32(false, a, false, b,
                                                          (short)0, c, false, false);
            }
            o[t] = c;
        }
    }

    #pragma unroll
    for (int t = 0; t < 12; ++t) {
        const int ncol  = wave * 192 + t * 16 + lrow;
        const float bias = b2g[ncol];
        #pragma unroll
        for (int j = 0; j < 8; ++j) {
            const int   mrow = 8 * half + j;
            const size_t idx = (size_t)(row0 + mrow) * DD + ncol;
            out[idx] = o[t][j] + bias + acc[idx];
        }
    }
}

// =====================================================================
// Kernel 4: balance-loss tail outputs.
// =====================================================================
__global__ __launch_bounds__(64) void finalize_kernel(
    const int* __restrict__ counter, const float* __restrict__ prob_sum,
    float* __restrict__ out_tail)
{
    __shared__ float part[EE];
    const int tid = threadIdx.x;           // 64 threads
    float freq = (float)counter[tid] * (1.0f / (float)(TT * KK));
    float prob = prob_sum[tid] * (1.0f / (float)TT);
    out_tail[1 + tid]      = freq;         // expert_freq
    out_tail[1 + EE + tid] = prob;         // expert_prob
    part[tid] = freq * prob;
    __syncthreads();
    if (tid == 0) {
        float s = 0.f;
        for (int i = 0; i < EE; ++i) s += part[i];
        out_tail[0] = 0.01f * s;           // balance_loss
    }
}

// =====================================================================
extern "C" void kernel_launch(void* const* d_in, const int* in_sizes, int n_in,
                              void* d_out, int out_size, void* d_ws, size_t ws_size,
                              hipStream_t stream) {
    (void)in_sizes; (void)n_in; (void)out_size; (void)ws_size;
    const float* x      = (const float*)d_in[0];
    const float* gate_w = (const float*)d_in[1];
    const float* sh1_w  = (const float*)d_in[2];
    const float* sh1_b  = (const float*)d_in[3];
    const float* sh2_w  = (const float*)d_in[4];
    const float* sh2_b  = (const float*)d_in[5];
    const float* e1_w   = (const float*)d_in[6];
    const float* e1_b   = (const float*)d_in[7];
    const float* e2_w   = (const float*)d_in[8];
    const float* e2_b   = (const float*)d_in[9];
    float* out = (float*)d_out;

    // workspace layout
    char* ws = (char*)d_ws;
    int*   counter   = (int*)ws;                               // 64 ints
    float* prob_sum  = (float*)(ws + 256);                     // 64 floats
    int*   route_tok = (int*)(ws + 512);                       // E*CAP ints
    float* route_sc  = (float*)(ws + 512 + (size_t)EE * CAPN * 4);
    float* acc       = (float*)(ws + (1 << 20));               // T*D floats @ 1MB

    hipMemsetAsync(ws, 0, 512, stream);                              // counters + prob sums
    hipMemsetAsync(acc, 0, (size_t)TT * DD * sizeof(float), stream); // routed accumulator

    gate_topk_kernel<<<TT / 8, 256, 0, stream>>>(
        x, gate_w, counter, prob_sum, route_tok, route_sc);

    expert_mlp_kernel<<<dim3(CAPN / 16, EE), 128, 0, stream>>>(
        x, e1_w, e1_b, e2_w, e2_b, counter, route_tok, route_sc, acc);

    shared_mlp_kernel<<<TT / 16, 128, 0, stream>>>(
        x, sh1_w, sh1_b, sh2_w, sh2_b, acc, out);

    finalize_kernel<<<1, 64, 0, stream>>>(
        counter, prob_sum, out + (size_t)TT * DD);
}